// CausalMessagePassingLayer_41807211659534
// MI455X (gfx1250) — compile-verified
//
#include <hip/hip_runtime.h>
#include <cstdint>
#include <cstddef>

#define B_   4
#define S_   8192
#define D_   768
#define NSUB 4096
#define EG   32768
#define LDA  772   // 768 + 4 floats padding -> conflict-free ds_load_b64 A-fragments

typedef float v2f __attribute__((ext_vector_type(2)));
typedef float v8f __attribute__((ext_vector_type(8)));

// ---------------- degree / normalization ----------------

__global__ void k_init_deg(float* __restrict__ deg) {
    int i = blockIdx.x * blockDim.x + threadIdx.x;   // exact multiple of 256
    deg[i] = 1.0f;                                   // self-loop contribution
}

__global__ void k_count_deg(const int* __restrict__ ei, float* __restrict__ deg) {
    int i = blockIdx.x * blockDim.x + threadIdx.x;   // over B*EG, exact
    int b = i / EG, e = i % EG;
    int dst = ei[((size_t)b * 2 + 1) * EG + e];
    __hip_atomic_fetch_add(&deg[b * NSUB + dst], 1.0f,
                           __ATOMIC_RELAXED, __HIP_MEMORY_SCOPE_AGENT);
}

__global__ void k_rsqrt(float* __restrict__ deg) {
    int i = blockIdx.x * blockDim.x + threadIdx.x;   // deg >= 1 always (self-loops)
    deg[i] = rsqrtf(deg[i]);
}

// ---------------- gathered GEMM: Y[b] = T[b][idx[b]] @ W  (f32 WMMA) ----------------

__global__ __launch_bounds__(256) void k_gemm_gather(
    const float* __restrict__ T, const int* __restrict__ idx,
    const float* __restrict__ W, float* __restrict__ Y) {
    __shared__ float lds[16 * LDA];

    const int b    = blockIdx.z;
    const int rb   = blockIdx.y;            // 16-row stripe of N_SUB
    const int cb   = blockIdx.x;            // 256-col block of D
    const int tid  = threadIdx.x;
    const int lane = tid & 31;
    const int wave = tid >> 5;

    // Cooperative gather of A stripe: 16 rows x 768 cols (float4-coalesced per row)
    const int rowBase = rb * 16;
#pragma unroll
    for (int i = 0; i < 12; ++i) {
        int j  = tid + i * 256;             // float4 index 0..3071
        int r  = j / 192;                   // A row 0..15
        int c4 = j % 192;                   // float4 column
        int srow = idx[b * NSUB + rowBase + r];
        float4 v = *(const float4*)(T + ((size_t)b * S_ + srow) * D_ + c4 * 4);
        *(float4*)(&lds[r * LDA + c4 * 4]) = v;
    }
    __syncthreads();

    const int m  = lane & 15;               // output row within tile / A row
    const int hi = lane >> 4;               // lane half selects K pair {0,1} vs {2,3}
    const int n0 = cb * 256 + wave * 32;    // wave's 32-col slab
    const int nc = n0 + m;                  // B/C column for this lane

    v8f acc0 = {};
    v8f acc1 = {};
    const float* lrow = &lds[m * LDA + 2 * hi];

#pragma unroll 4
    for (int k = 0; k < D_; k += 4) {
        v2f a;
        a.x = lrow[k];
        a.y = lrow[k + 1];
        const float* wr = W + (size_t)(k + 2 * hi) * D_;
        v2f b0; b0.x = wr[nc];      b0.y = wr[D_ + nc];
        v2f b1; b1.x = wr[nc + 16]; b1.y = wr[D_ + nc + 16];
        acc0 = __builtin_amdgcn_wmma_f32_16x16x4_f32(false, a, false, b0,
                                                     (short)0, acc0, false, false);
        acc1 = __builtin_amdgcn_wmma_f32_16x16x4_f32(false, a, false, b1,
                                                     (short)0, acc1, false, false);
    }

    // C/D layout: VGPR v, lo lanes -> row v, hi lanes -> row v+8; col = n0 + (lane&15)
#pragma unroll
    for (int v = 0; v < 8; ++v) {
        int row = rowBase + v + hi * 8;
        float* yr = Y + ((size_t)b * NSUB + row) * D_ + n0 + m;
        yr[0]  = acc0[v];
        yr[16] = acc1[v];
    }
}

// ---------------- self-loop + bias: agg = dinv^2 * xw + bias ----------------

__global__ void k_agg_init(const float* __restrict__ xw, const float* __restrict__ dinv,
                           const float* __restrict__ bias, float* __restrict__ agg) {
    int i   = blockIdx.x * blockDim.x + threadIdx.x; // float4 idx over B*NSUB*192, exact
    int row = i / 192;
    int d4  = i % 192;
    float s = dinv[row]; s = s * s;
    float4 x = *(const float4*)(xw + (size_t)i * 4);
    float4 bb = *(const float4*)(bias + d4 * 4);
    float4 o;
    o.x = s * x.x + bb.x;
    o.y = s * x.y + bb.y;
    o.z = s * x.z + bb.z;
    o.w = s * x.w + bb.w;
    *(float4*)(agg + (size_t)i * 4) = o;
}

// ---------------- edge aggregation: agg[dst] += dinv[src]*dinv[dst]*xw[src] ----------------

__global__ __launch_bounds__(256) void k_agg_edges(
    const int* __restrict__ ei, const float* __restrict__ dinv,
    const float* __restrict__ xw, float* __restrict__ agg) {
    int w    = blockIdx.x * 8 + (threadIdx.x >> 5);  // global wave = edge over B*EG, exact
    int lane = threadIdx.x & 31;
    int b = w / EG, e = w % EG;
    const int* eib = ei + (size_t)b * 2 * EG;
    int src = eib[e];
    int dst = eib[EG + e];
    float norm = dinv[b * NSUB + src] * dinv[b * NSUB + dst];
    const float* xs = xw  + ((size_t)b * NSUB + src) * D_;
    float*       ad = agg + ((size_t)b * NSUB + dst) * D_;
#pragma unroll
    for (int i = 0; i < 6; ++i) {
        int d = (lane + i * 32) * 4;
        float4 v = *(const float4*)(xs + d);
        __hip_atomic_fetch_add(ad + d + 0, norm * v.x, __ATOMIC_RELAXED, __HIP_MEMORY_SCOPE_AGENT);
        __hip_atomic_fetch_add(ad + d + 1, norm * v.y, __ATOMIC_RELAXED, __HIP_MEMORY_SCOPE_AGENT);
        __hip_atomic_fetch_add(ad + d + 2, norm * v.z, __ATOMIC_RELAXED, __HIP_MEMORY_SCOPE_AGENT);
        __hip_atomic_fetch_add(ad + d + 3, norm * v.w, __ATOMIC_RELAXED, __HIP_MEMORY_SCOPE_AGENT);
    }
}

// ---------------- scatter: out[b, map[n], :] += tanh(gate) * agg[b, n, :] ----------------

__global__ __launch_bounds__(256) void k_scatter(
    const float* __restrict__ agg, const int* __restrict__ map,
    const float* __restrict__ gate, float* __restrict__ out) {
    int w    = blockIdx.x * 8 + (threadIdx.x >> 5);  // n over B*NSUB, exact
    int lane = threadIdx.x & 31;
    int b = w / NSUB, n = w % NSUB;
    float g = tanhf(gate[0]);
    int s = map[b * NSUB + n];
    const float* ar = agg + ((size_t)b * NSUB + n) * D_;
    float*     orow = out + ((size_t)b * S_   + s) * D_;
#pragma unroll
    for (int i = 0; i < 6; ++i) {
        int d = (lane + i * 32) * 4;
        float4 a = *(const float4*)(ar + d);
        float4 o = *(const float4*)(orow + d);
        o.x += g * a.x;
        o.y += g * a.y;
        o.z += g * a.z;
        o.w += g * a.w;
        *(float4*)(orow + d) = o;
    }
}

// ---------------- host launch ----------------

extern "C" void kernel_launch(void* const* d_in, const int* in_sizes, int n_in,
                              void* d_out, int out_size, void* d_ws, size_t ws_size,
                              hipStream_t stream) {
    const float* T   = (const float*)d_in[0];
    const int*   t2e = (const int*)d_in[1];
    const int*   e2t = (const int*)d_in[2];
    const int*   t2n = (const int*)d_in[3];
    const int*   n2t = (const int*)d_in[4];
    const int*   eiE = (const int*)d_in[5];
    const int*   eiN = (const int*)d_in[6];
    const float* W_e = (const float*)d_in[7];
    const float* b_e = (const float*)d_in[8];
    const float* W_n = (const float*)d_in[9];
    const float* b_n = (const float*)d_in[10];
    const float* ga  = (const float*)d_in[11];
    const float* gb  = (const float*)d_in[12];
    float* out = (float*)d_out;

    char* ws = (char*)d_ws;
    float* deg = (float*)ws;                                        // B*NSUB floats
    float* xw  = (float*)(ws + 256 * 1024);                         // B*NSUB*D floats
    float* agg = (float*)(ws + 256 * 1024 +
                          (size_t)B_ * NSUB * D_ * sizeof(float));  // B*NSUB*D floats

    // out = token_embeddings
    hipMemcpyAsync(d_out, d_in[0], (size_t)B_ * S_ * D_ * sizeof(float),
                   hipMemcpyDeviceToDevice, stream);

    const int*   idxs[2]  = {t2e, t2n};
    const int*   maps[2]  = {e2t, n2t};
    const int*   eis[2]   = {eiE, eiN};
    const float* Ws[2]    = {W_e, W_n};
    const float* bs[2]    = {b_e, b_n};
    const float* gates[2] = {ga, gb};

    for (int g = 0; g < 2; ++g) {
        k_init_deg <<<(B_ * NSUB) / 256, 256, 0, stream>>>(deg);
        k_count_deg<<<(B_ * EG)   / 256, 256, 0, stream>>>(eis[g], deg);
        k_rsqrt    <<<(B_ * NSUB) / 256, 256, 0, stream>>>(deg);
        k_gemm_gather<<<dim3(3, NSUB / 16, B_), 256, 0, stream>>>(T, idxs[g], Ws[g], xw);
        k_agg_init <<<(B_ * NSUB * (D_ / 4)) / 256, 256, 0, stream>>>(xw, deg, bs[g], agg);
        k_agg_edges<<<(B_ * EG) / 8, 256, 0, stream>>>(eis[g], deg, xw, agg);
        k_scatter  <<<(B_ * NSUB) / 8, 256, 0, stream>>>(agg, maps[g], gates[g], out);
    }
}